// Inference_and_Generation_54975581389075
// MI455X (gfx1250) — compile-verified
//
#include <hip/hip_runtime.h>
#include <math.h>

typedef float v2f __attribute__((ext_vector_type(2)));
typedef float v8f __attribute__((ext_vector_type(8)));
typedef unsigned int u32x4 __attribute__((ext_vector_type(4)));
typedef int i32x4 __attribute__((ext_vector_type(4)));
typedef int i32x8 __attribute__((ext_vector_type(8)));

#if defined(__has_builtin)
#if __has_builtin(__builtin_amdgcn_tensor_load_to_lds) && __has_builtin(__builtin_amdgcn_s_wait_tensorcnt)
#define HAVE_TDM 1
#endif
#endif
#ifndef HAVE_TDM
#define HAVE_TDM 0
#endif

// ---------------- problem sizes ----------------
constexpr int NB    = 8;      // batch
constexpr int KOBJ  = 48;
constexpr int KB    = KOBJ * NB;       // 384 rows of the GEMMs
constexpr int CF    = 32;
constexpr int WRAW  = 128;
constexpr int CROPS = 28;
constexpr int FENC  = CF * CROPS * CROPS;   // 25088
constexpr int FDEC  = 2 * CROPS * CROPS;    // 1568
constexpr int DENC  = 64;                   // 2*ZI
constexpr int ZIDIM = 32;
constexpr int SPLITK = 16;
constexpr int KCHUNK = FENC / SPLITK;       // 1568
constexpr int KSTEPS = KCHUNK / 4;          // 392
constexpr int TDM_TILEF = 32;               // floats of K per TDM tile
constexpr int NTILES = KCHUNK / TDM_TILEF;  // 49
constexpr int LDS_STRIDE = TDM_TILEF + 1;   // 33: TDM pad -> conflict-free banks

// ---------------- output offsets (floats, return-tuple order) ----------------
constexpr int O_PROBMAP    = 0;
constexpr int O_AREAMAP    = 8192;
constexpr int O_CFEW       = 16384;
constexpr int O_PROBFEW    = 16768;
constexpr int O_BX         = 17152;
constexpr int O_BY         = 17536;
constexpr int O_BW         = 17920;
constexpr int O_BH         = 18304;
constexpr int O_BIGMASK    = 18688;
constexpr int O_BIGMASKNON = 6310144;
constexpr int O_BIGIMG     = 12601600;
constexpr int O_ZWS        = 18893056;
constexpr int O_ZWKL       = 18896128;
constexpr int O_ZIS        = 18899200;
constexpr int O_ZIKL       = 18911488;

// ---------------- workspace offsets (floats) ----------------
constexpr int W_BX    = 0;        // 8192 each, layout [n*8+b]
constexpr int W_BY    = 8192;
constexpr int W_BW    = 16384;
constexpr int W_BH    = 24576;
constexpr int W_IDX   = 32768;    // int[48*8], layout [k*8+b]
constexpr int W_PART  = 33152;    // split-K partials: [kc][kb][d] 16*384*64
constexpr int W_ZI    = 426368;   // zi_sample [kb][32]
constexpr int W_SMALL = 438656;   // small*c   [kb][1568]

// ---------------- helpers ----------------
__device__ __forceinline__ float sigmoidf_(float x) { return 1.0f / (1.0f + __expf(-x)); }
__device__ __forceinline__ float softplusf_(float x) {
    return (x > 20.0f) ? x : log1pf(expf(x));
}
__device__ __forceinline__ int clampi(int v, int lo, int hi) {
    return v < lo ? lo : (v > hi ? hi : v);
}

__device__ __forceinline__ float fetch2d(const float* img, int xi, int yi, int W, int H) {
    bool ok = ((unsigned)xi < (unsigned)W) && ((unsigned)yi < (unsigned)H);
    float v = img[clampi(xi, 0, W - 1) * H + clampi(yi, 0, H - 1)];
    return ok ? v : 0.0f;
}

__device__ __forceinline__ float bilinear(const float* img, float xs, float ys, int W, int H) {
    float x0f = floorf(xs), y0f = floorf(ys);
    int   x0  = (int)x0f,   y0  = (int)y0f;
    float wx1 = xs - x0f, wx0 = 1.0f - wx1;
    float wy1 = ys - y0f, wy0 = 1.0f - wy1;
    float v00 = fetch2d(img, x0,     y0,     W, H);
    float v10 = fetch2d(img, x0 + 1, y0,     W, H);
    float v01 = fetch2d(img, x0,     y0 + 1, W, H);
    float v11 = fetch2d(img, x0 + 1, y0 + 1, W, H);
    return v00 * (wx0 * wy0) + v10 * (wx1 * wy0) + v01 * (wx0 * wy1) + v11 * (wx1 * wy1);
}

// ===================================================================
// K1: prob_map, area_map, box-parameter lists
// ===================================================================
__global__ void prep_kernel(const float* __restrict__ logit,
                            const float* __restrict__ zmu,
                            const float* __restrict__ zstd,
                            const float* __restrict__ zeps,
                            const float* __restrict__ wz,
                            const float* __restrict__ bz,
                            float* __restrict__ out, float* __restrict__ ws) {
    int t = blockIdx.x * blockDim.x + threadIdx.x;     // 0..8191  = b*1024 + n
    int b = t >> 10, n = t & 1023;
    int w = n >> 5,  h = n & 31;
    out[O_PROBMAP + t] = sigmoidf_(logit[t]);

    float zs[8];
#pragma unroll
    for (int z = 0; z < 8; ++z) {
        int off = b * 8192 + z * 1024 + n;
        zs[z] = zmu[off] + zstd[off] * zeps[off];
    }
    float tm[4];
#pragma unroll
    for (int c = 0; c < 4; ++c) {
        float a = bz[c];
#pragma unroll
        for (int z = 0; z < 8; ++z) a += zs[z] * wz[c * 8 + z];
        tm[c] = sigmoidf_(a);
    }
    float bx = 4.0f * ((float)w + tm[0]);              // 128*(w+tx)/32
    float by = 4.0f * ((float)h + tm[1]);
    float bw = 10.0f + 30.0f * tm[2];
    float bh = 10.0f + 30.0f * tm[3];
    out[O_AREAMAP + t] = bw * bh;
    ws[W_BX + n * 8 + b] = bx;
    ws[W_BY + n * 8 + b] = by;
    ws[W_BW + n * 8 + b] = bw;
    ws[W_BH + n * 8 + b] = bh;
}

// ===================================================================
// K2: top-48 per batch, exact jax.lax.top_k tie-breaking (lower index wins)
// ===================================================================
__global__ void topk_kernel(const float* __restrict__ logit, float* __restrict__ ws) {
    __shared__ unsigned long long keys[1024];
    __shared__ unsigned long long red[256];
    int b = blockIdx.x, tid = threadIdx.x;
    for (int n = tid; n < 1024; n += 256) {
        unsigned u = __float_as_uint(logit[b * 1024 + n]);
        u = (u & 0x80000000u) ? ~u : (u | 0x80000000u);   // monotone float order
        keys[n] = ((unsigned long long)u << 32) | (unsigned)(1023 - n);
    }
    __syncthreads();
    int* idx = (int*)(ws + W_IDX);
    for (int r = 0; r < KOBJ; ++r) {
        unsigned long long best = 0;
        for (int n = tid; n < 1024; n += 256) {
            unsigned long long k = keys[n];
            best = (k > best) ? k : best;
        }
        red[tid] = best;
        __syncthreads();
        for (int off = 128; off > 0; off >>= 1) {
            if (tid < off) { unsigned long long o = red[tid + off]; if (o > red[tid]) red[tid] = o; }
            __syncthreads();
        }
        if (tid == 0) {
            int n = 1023 - (int)(red[0] & 0xFFFFFFFFull);
            idx[r * 8 + b] = n;
            keys[n] = 0;
        }
        __syncthreads();
    }
}

// ===================================================================
// K3: gather few-tensors + zwhere sample/KL (recomputed)
// ===================================================================
__global__ void gather_kernel(const float* __restrict__ logit,
                              const float* __restrict__ zmu,
                              const float* __restrict__ zstd,
                              const float* __restrict__ zeps,
                              float* __restrict__ out, const float* __restrict__ ws) {
    int kb = blockIdx.x * blockDim.x + threadIdx.x;
    if (kb >= KB) return;
    int k = kb >> 3, b = kb & 7;
    const int* idx = (const int*)(ws + W_IDX);
    int n = idx[k * 8 + b];
    float lg = logit[b * 1024 + n];
    out[O_CFEW + kb]    = lg;
    out[O_PROBFEW + kb] = sigmoidf_(lg);
    out[O_BX + kb] = ws[W_BX + n * 8 + b];
    out[O_BY + kb] = ws[W_BY + n * 8 + b];
    out[O_BW + kb] = ws[W_BW + n * 8 + b];
    out[O_BH + kb] = ws[W_BH + n * 8 + b];
#pragma unroll
    for (int z = 0; z < 8; ++z) {
        int off = b * 8192 + z * 1024 + n;
        float mu = zmu[off], sd = zstd[off];
        float s  = mu + sd * zeps[off];
        out[O_ZWS  + kb * 8 + z] = s;
        out[O_ZWKL + kb * 8 + z] = 0.5f * (mu * mu + sd * sd) - logf(sd) - 0.5f;
    }
}

// ===================================================================
// TDM: async-load a 64-row x 32-float tile of w_enc into LDS.
//   D# group0: count=1, lds_addr, global_addr, type=2
//   D# group1: data_size=4B, pad_enable, pad_interval=128B, pad_amount=1dw
//              tensor_dim0=FENC, tensor_dim1=64, tile=32x64,
//              tensor_dim0_stride=FENC  -> LDS row stride 33 dwords
// ===================================================================
#if HAVE_TDM
__device__ __forceinline__ void tdm_load_btile(const float* gptr, void* lds_ptr) {
    unsigned long long ga = (unsigned long long)(size_t)gptr;
    unsigned lds = (unsigned)(size_t)lds_ptr;     // low 32 bits = LDS byte offset
    u32x4 g0;
    g0[0] = 1u;                                   // count=1, user descriptor
    g0[1] = lds;
    g0[2] = (unsigned)(ga & 0xFFFFFFFFull);
    g0[3] = (unsigned)((ga >> 32) & 0x01FFFFFFull) | (2u << 30);   // type=2 (image)
    i32x8 g1;
    unsigned td0 = (unsigned)FENC;                // tensor_dim0
    unsigned td1 = 64u;                           // tensor_dim1 (d rows)
    unsigned long long s0 = (unsigned long long)FENC;   // tensor_dim0_stride
    g1[0] = (int)((2u << 16) | (1u << 20) | (4u << 22));  // 4B elems, pad: every 32dw +1dw
    g1[1] = (int)((td0 & 0xFFFFu) << 16);                 // [31:16]=dim0 lo
    g1[2] = (int)((td0 >> 16) | ((td1 & 0xFFFFu) << 16)); // dim0 hi | dim1 lo
    g1[3] = (int)((td1 >> 16) | ((unsigned)TDM_TILEF << 16)); // dim1 hi | tile_dim0
    g1[4] = (int)64u;                                     // tile_dim1=64, tile_dim2=0
    g1[5] = (int)(s0 & 0xFFFFFFFFull);                    // dim0_stride lo
    g1[6] = (int)((s0 >> 32) & 0xFFFFull);                // dim0_stride hi | dim1_stride lo
    g1[7] = 0;
    i32x4 z4 = {0, 0, 0, 0};
#if __clang_major__ >= 23
    i32x8 z8 = {0, 0, 0, 0, 0, 0, 0, 0};
    __builtin_amdgcn_tensor_load_to_lds(g0, g1, z4, z4, z8, 0);
#else
    __builtin_amdgcn_tensor_load_to_lds(g0, g1, z4, z4, 0);
#endif
}
#endif

// ===================================================================
// K4: encoder GEMM with fused bilinear crop.
//   D[kb,d] = sum_f crop(kb,f) * w_enc[d,f]   (M=384, N=64, K=25088)
//   One wave per (M-tile, K-chunk); 4 accumulators cover N=64.
//   w_enc tiles streamed by the TDM into double-buffered LDS while the
//   VALU computes the bilinear A fragments; V_WMMA_F32_16X16X4_F32.
// ===================================================================
__device__ __forceinline__ float crop_sample(const float* __restrict__ img_b, int f,
                                             float xb, float yb, float sx, float sy) {
    int c = f / 784;
    int r = f - c * 784;
    int i = r / 28;
    int j = r - i * 28;
    float xs = xb + (float)i * sx;
    float ys = yb + (float)j * sy;
    return bilinear(img_b + c * 16384, xs, ys, WRAW, WRAW);
}

__device__ __forceinline__ void enc_consume_tile(
        const float* __restrict__ Bt, int kbase, int koff, int row,
        const float* __restrict__ img_b, float xb, float yb, float sx, float sy,
        v8f& acc0, v8f& acc1, v8f& acc2, v8f& acc3) {
#pragma unroll
    for (int s8 = 0; s8 < 8; ++s8) {
        const int f0 = kbase + s8 * 4 + koff;
        v2f a;
        a.x = crop_sample(img_b, f0,     xb, yb, sx, sy);
        a.y = crop_sample(img_b, f0 + 1, xb, yb, sx, sy);
        const int lf = s8 * 4 + koff;
        v2f b0, b1, b2, b3;
        b0.x = Bt[(0 * 16 + row) * LDS_STRIDE + lf]; b0.y = Bt[(0 * 16 + row) * LDS_STRIDE + lf + 1];
        b1.x = Bt[(1 * 16 + row) * LDS_STRIDE + lf]; b1.y = Bt[(1 * 16 + row) * LDS_STRIDE + lf + 1];
        b2.x = Bt[(2 * 16 + row) * LDS_STRIDE + lf]; b2.y = Bt[(2 * 16 + row) * LDS_STRIDE + lf + 1];
        b3.x = Bt[(3 * 16 + row) * LDS_STRIDE + lf]; b3.y = Bt[(3 * 16 + row) * LDS_STRIDE + lf + 1];
        acc0 = __builtin_amdgcn_wmma_f32_16x16x4_f32(false, a, false, b0, (short)0, acc0, false, false);
        acc1 = __builtin_amdgcn_wmma_f32_16x16x4_f32(false, a, false, b1, (short)0, acc1, false, false);
        acc2 = __builtin_amdgcn_wmma_f32_16x16x4_f32(false, a, false, b2, (short)0, acc2, false, false);
        acc3 = __builtin_amdgcn_wmma_f32_16x16x4_f32(false, a, false, b3, (short)0, acc3, false, false);
    }
}

__global__ void __launch_bounds__(32) enc_gemm_kernel(const float* __restrict__ feat,
                                                      const float* __restrict__ wenc,
                                                      const float* __restrict__ out,
                                                      float* __restrict__ ws) {
    const int mt   = blockIdx.x;           // 0..23
    const int kc   = blockIdx.y;           // 0..15
    const int lane = threadIdx.x;
    const int m0   = mt * 16;
    const int row  = lane & 15;            // A row / B column within tile
    const int koff = (lane >> 4) << 1;     // lanes 0-15 -> K 0,1 ; lanes 16-31 -> K 2,3
    const int m    = m0 + row;
    const int b    = m & 7;

    // per-row box parameters are loop-invariant for this lane
    const float bx = out[O_BX + m], by = out[O_BY + m];
    const float bw = out[O_BW + m], bh = out[O_BH + m];
    const float sx = bw * (1.0f / 28.0f), sy = bh * (1.0f / 28.0f);
    const float xb = bx - 0.5f * bw + 0.5f * sx - 0.5f;
    const float yb = by - 0.5f * bh + 0.5f * sy - 0.5f;
    const float* img_b = feat + (size_t)b * CF * 16384;

    v8f acc0 = {}, acc1 = {}, acc2 = {}, acc3 = {};
    const int fbase = kc * KCHUNK;

#if HAVE_TDM
    __shared__ float ldsB[2][64 * LDS_STRIDE];
    tdm_load_btile(wenc + fbase, &ldsB[0][0]);
    for (int tile = 0; tile < NTILES - 1; ++tile) {
        tdm_load_btile(wenc + fbase + (tile + 1) * TDM_TILEF, &ldsB[(tile + 1) & 1][0]);
        __builtin_amdgcn_s_wait_tensorcnt(1);        // current tile resident
        enc_consume_tile(&ldsB[tile & 1][0], fbase + tile * TDM_TILEF, koff, row,
                         img_b, xb, yb, sx, sy, acc0, acc1, acc2, acc3);
    }
    __builtin_amdgcn_s_wait_tensorcnt(0);
    enc_consume_tile(&ldsB[(NTILES - 1) & 1][0], fbase + (NTILES - 1) * TDM_TILEF, koff, row,
                     img_b, xb, yb, sx, sy, acc0, acc1, acc2, acc3);
#else
    for (int s = 0; s < KSTEPS; ++s) {
        const int kk = fbase + s * 4;
        const int f0 = kk + koff;
        v2f a;
        a.x = crop_sample(img_b, f0,     xb, yb, sx, sy);
        a.y = crop_sample(img_b, f0 + 1, xb, yb, sx, sy);
        const float* wp = wenc + (size_t)(kk + koff);
        __builtin_prefetch(wp + (size_t)row * FENC + 64, 0, 1);
        v2f b0, b1, b2, b3;
        b0.x = wp[(0 * 16 + row) * FENC]; b0.y = wp[(0 * 16 + row) * FENC + 1];
        b1.x = wp[(1 * 16 + row) * FENC]; b1.y = wp[(1 * 16 + row) * FENC + 1];
        b2.x = wp[(2 * 16 + row) * FENC]; b2.y = wp[(2 * 16 + row) * FENC + 1];
        b3.x = wp[(3 * 16 + row) * FENC]; b3.y = wp[(3 * 16 + row) * FENC + 1];
        acc0 = __builtin_amdgcn_wmma_f32_16x16x4_f32(false, a, false, b0, (short)0, acc0, false, false);
        acc1 = __builtin_amdgcn_wmma_f32_16x16x4_f32(false, a, false, b1, (short)0, acc1, false, false);
        acc2 = __builtin_amdgcn_wmma_f32_16x16x4_f32(false, a, false, b2, (short)0, acc2, false, false);
        acc3 = __builtin_amdgcn_wmma_f32_16x16x4_f32(false, a, false, b3, (short)0, acc3, false, false);
    }
#endif

    float* part = ws + W_PART + (size_t)kc * KB * DENC;
    const int rbase = m0 + ((lane >> 4) << 3);        // rows v (lanes<16) / v+8 (lanes>=16)
#pragma unroll
    for (int v = 0; v < 8; ++v) {
        const int rm = (rbase + v) * DENC;
        part[rm + 0 * 16 + row] = acc0[v];
        part[rm + 1 * 16 + row] = acc1[v];
        part[rm + 2 * 16 + row] = acc2[v];
        part[rm + 3 * 16 + row] = acc3[v];
    }
}

// ===================================================================
// K5: deterministic split-K reduce + bias + reparameterize z_instance
// ===================================================================
__global__ void zi_kernel(const float* __restrict__ benc,
                          const float* __restrict__ zieps,
                          float* __restrict__ out, float* __restrict__ ws) {
    int t  = blockIdx.x * blockDim.x + threadIdx.x;    // 0..12287 = kb*32+z
    int kb = t >> 5, z = t & 31;
    float mu = benc[z], sp = benc[z + 32];
    const float* part = ws + W_PART;
#pragma unroll
    for (int kc = 0; kc < SPLITK; ++kc) {
        const float* p = part + (size_t)kc * KB * DENC + kb * DENC;
        mu += p[z];
        sp += p[z + 32];
    }
    float sd = softplusf_(sp) + 1e-4f;
    float s  = mu + sd * zieps[t];
    out[O_ZIS  + t] = s;
    out[O_ZIKL + t] = 0.5f * (mu * mu + sd * sd) - logf(sd) - 0.5f;
    ws[W_ZI + t] = s;
}

// ===================================================================
// K6: decoder GEMM (M=384, N=1568, K=32) + fused softplus/sigmoid * c_few
// ===================================================================
__global__ void __launch_bounds__(32) dec_gemm_kernel(const float* __restrict__ wdec,
                                                      const float* __restrict__ bdec,
                                                      const float* __restrict__ out,
                                                      float* __restrict__ ws) {
    const int nt   = blockIdx.x;     // 0..97
    const int mt   = blockIdx.y;     // 0..23
    const int lane = threadIdx.x;
    const int m0 = mt * 16, n0 = nt * 16;
    const int row  = lane & 15;
    const int koff = (lane >> 4) << 1;
    const float* zi = ws + W_ZI;

    v8f acc = {};
#pragma unroll
    for (int s = 0; s < 8; ++s) {
        const int kk = s * 4 + koff;
        v2f a;  a.x  = zi[(m0 + row) * ZIDIM + kk];   a.y  = zi[(m0 + row) * ZIDIM + kk + 1];
        v2f bb; bb.x = wdec[(n0 + row) * ZIDIM + kk]; bb.y = wdec[(n0 + row) * ZIDIM + kk + 1];
        acc = __builtin_amdgcn_wmma_f32_16x16x4_f32(false, a, false, bb, (short)0, acc, false, false);
    }

    const int rbase = m0 + ((lane >> 4) << 3);
    const int f     = n0 + row;
    const float bias = bdec[f];
#pragma unroll
    for (int v = 0; v < 8; ++v) {
        const int rm = rbase + v;
        float r   = acc[v] + bias;
        float val = (f >= 784) ? sigmoidf_(r) : softplusf_(r);   // ch1 sigmoid, ch0 softplus
        ws[W_SMALL + rm * FDEC + f] = val * out[O_CFEW + rm];
    }
}

// ===================================================================
// K7: un-crop (bilinear resample small -> 128x128); big_img direct,
//     big_weight staged into the big_mask output slot (saves 25MB of ws)
// ===================================================================
__global__ void uncrop_kernel(float* __restrict__ out, const float* __restrict__ ws) {
    const int kb = blockIdx.y;
    const int p  = blockIdx.x * blockDim.x + threadIdx.x;   // 0..16383
    const int X = p >> 7, Y = p & 127;
    const float bx = out[O_BX + kb], by = out[O_BY + kb];
    const float bw = out[O_BW + kb], bh = out[O_BH + kb];
    const float xs = ((float)X + 0.5f - (bx - 0.5f * bw)) / bw * 28.0f - 0.5f;
    const float ys = ((float)Y + 0.5f - (by - 0.5f * bh)) / bh * 28.0f - 0.5f;
    const float* sm = ws + W_SMALL + (size_t)kb * FDEC;
    float wgt = bilinear(sm,       xs, ys, CROPS, CROPS);   // channel 0
    float im  = bilinear(sm + 784, xs, ys, CROPS, CROPS);   // channel 1
    out[O_BIGMASK + (size_t)kb * 16384 + p] = wgt;          // temp: big_weight
    out[O_BIGIMG  + (size_t)kb * 16384 + p] = im;
}

// ===================================================================
// K8: per-pixel reduce over k, then big_mask / big_mask_non
// ===================================================================
__global__ void mask_kernel(float* __restrict__ out) {
    const int t = blockIdx.x * blockDim.x + threadIdx.x;    // 0..131071 = b*16384+p
    const int b = t >> 14, p = t & 16383;
    float wv[KOBJ];
    float s = 0.0f;
#pragma unroll
    for (int k = 0; k < KOBJ; ++k) {
        wv[k] = out[O_BIGMASK + (size_t)(k * 8 + b) * 16384 + p];
        s += wv[k];
    }
    const float scale = tanhf(s) / fmaxf(s, 1e-6f);
#pragma unroll
    for (int k = 0; k < KOBJ; ++k) {
        const size_t o = (size_t)(k * 8 + b) * 16384 + p;
        out[O_BIGMASK    + o] = wv[k] * scale;
        out[O_BIGMASKNON + o] = tanhf(wv[k]);
    }
}

// ===================================================================
extern "C" void kernel_launch(void* const* d_in, const int* in_sizes, int n_in,
                              void* d_out, int out_size, void* d_ws, size_t ws_size,
                              hipStream_t stream) {
    const float* logit = (const float*)d_in[0];
    const float* zmu   = (const float*)d_in[1];
    const float* zstd  = (const float*)d_in[2];
    const float* zeps  = (const float*)d_in[3];
    const float* zieps = (const float*)d_in[4];
    const float* feat  = (const float*)d_in[5];
    const float* wz    = (const float*)d_in[6];
    const float* bz    = (const float*)d_in[7];
    const float* wenc  = (const float*)d_in[8];
    const float* benc  = (const float*)d_in[9];
    const float* wdec  = (const float*)d_in[10];
    const float* bdec  = (const float*)d_in[11];
    float* out = (float*)d_out;
    float* ws  = (float*)d_ws;

    prep_kernel<<<32, 256, 0, stream>>>(logit, zmu, zstd, zeps, wz, bz, out, ws);
    topk_kernel<<<8, 256, 0, stream>>>(logit, ws);
    gather_kernel<<<3, 128, 0, stream>>>(logit, zmu, zstd, zeps, out, ws);
    enc_gemm_kernel<<<dim3(24, 16), 32, 0, stream>>>(feat, wenc, out, ws);
    zi_kernel<<<48, 256, 0, stream>>>(benc, zieps, out, ws);
    dec_gemm_kernel<<<dim3(98, 24), 32, 0, stream>>>(wdec, bdec, out, ws);
    uncrop_kernel<<<dim3(64, KB), 256, 0, stream>>>(out, ws);
    mask_kernel<<<512, 256, 0, stream>>>(out);
}